// HNN_43379169689793
// MI455X (gfx1250) — compile-verified
//
#include <hip/hip_runtime.h>

typedef __attribute__((ext_vector_type(2))) float v2f;
typedef __attribute__((ext_vector_type(8))) float v8f;

#define HH 256  // hidden dim

// ---- WMMA f32 16x16x4 wrapper (CDNA5 / gfx1250) -----------------------------
// Layouts per CDNA5 ISA 7.12.2:
//   A 16x4 (2 VGPRs/lane): lane l holds row M=l%16, VGPR0=K(kOff), VGPR1=K(kOff+1),
//     kOff = (lane<16) ? 0 : 2.
//   B 4x16 (2 VGPRs/lane): lane l holds col N=l%16, rows K=kOff, kOff+1.
//   C/D 16x16 f32 (8 VGPRs): VGPR r = (M=r, N=lane) for lanes 0-15,
//     (M=r+8, N=lane-16) for lanes 16-31.
__device__ __forceinline__ v8f wmma4(v2f a, v2f b, v8f c) {
  return __builtin_amdgcn_wmma_f32_16x16x4_f32(
      /*neg_a=*/false, a, /*neg_b=*/false, b,
      /*c_mod=*/(short)0, c, /*reuse_a=*/false, /*reuse_b=*/false);
}

__device__ __forceinline__ v8f v8splat(float x) {
  v8f r;
#pragma unroll
  for (int i = 0; i < 8; ++i) r[i] = x;
  return r;
}

// ---- degrees ----------------------------------------------------------------
__global__ void k_deg(const int* __restrict__ src, const int* __restrict__ dst,
                      float* __restrict__ outdeg, float* __restrict__ indeg, int E) {
  int e = blockIdx.x * blockDim.x + threadIdx.x;
  if (e < E) {
    atomicAdd(&outdeg[src[e]], 1.0f);
    atomicAdd(&indeg[dst[e]], 1.0f);
  }
}

// ---- per-node prep: norms, mass diagonal, q*norm_src, dHdP ------------------
__global__ void k_prep(const float* __restrict__ q, const float* __restrict__ p,
                       const float* __restrict__ M,
                       const float* __restrict__ outdeg, const float* __restrict__ indeg,
                       float* __restrict__ ns, float* __restrict__ nd,
                       float* __restrict__ mdg, float* __restrict__ xs,
                       float* __restrict__ out, int n) {
  int i = blockIdx.x * blockDim.x + threadIdx.x;
  if (i >= n) return;
  float od = outdeg[i]; od = od > 0.f ? od : 1.f;
  float ig = indeg[i];  ig = ig > 0.f ? ig : 1.f;
  float s = 1.0f / sqrtf(od);
  float t = 1.0f / sqrtf(ig);
  ns[i] = s;
  nd[i] = t;
#pragma unroll
  for (int d = 0; d < 8; ++d) {
    float m = M[(size_t)i * 64 + d * 9];     // diagonal of M_i
    mdg[(size_t)i * 8 + d] = m;
    xs[(size_t)i * 8 + d] = q[(size_t)i * 8 + d] * s;
    out[(size_t)i * 16 + 8 + d] = p[(size_t)i * 8 + d] / m;  // dHdP = M^{-T} p
  }
}

// ---- zero-padded weight tables: W2p[h][n<16], W1tp[h][n<16] -----------------
// W2p[h*16+n]  = (n<8) ? W2[h*8+n]   : 0     (h1 @ W2 B-operand, K rows = h)
// W1tp[h*16+n] = (n<8) ? W1[n*256+h] : 0     (u  @ W1^T B-operand, K rows = h)
__global__ void k_padw(const float* __restrict__ W1, const float* __restrict__ W2,
                       float* __restrict__ W2p, float* __restrict__ W1tp) {
  int i = blockIdx.x * blockDim.x + threadIdx.x;   // 0 .. 8191
  int h = (i >> 4) & 255;
  int n = i & 15;
  if (i < 4096) {
    W2p[i] = (n < 8) ? W2[h * 8 + n] : 0.0f;
  } else if (i < 8192) {
    W1tp[i - 4096] = (n < 8) ? W1[n * HH + h] : 0.0f;
  }
}

// ---- generic edge gather/scatter over 8 features: acc[si] += x[gi]*(sc?sc[gi]:1)
__global__ void k_edge_agg(const float* __restrict__ x, const int* __restrict__ gi,
                           const int* __restrict__ si, const float* __restrict__ sc,
                           float* __restrict__ acc, int E) {
  int e = blockIdx.x * blockDim.x + threadIdx.x;
  if (e >= E) return;
  int g = gi[e], s = si[e];
  const float4* xp = (const float4*)(x + (size_t)g * 8);
  float4 v0 = xp[0], v1 = xp[1];
  float k = sc ? sc[g] : 1.0f;
  float* a = acc + (size_t)s * 8;
  atomicAdd(a + 0, v0.x * k); atomicAdd(a + 1, v0.y * k);
  atomicAdd(a + 2, v0.z * k); atomicAdd(a + 3, v0.w * k);
  atomicAdd(a + 4, v1.x * k); atomicAdd(a + 5, v1.y * k);
  atomicAdd(a + 6, v1.z * k); atomicAdd(a + 7, v1.w * k);
}

// ---- forward fused: ys = (relu((A q) W1 + b1) W2) * norm_src ----------------
// one wave per 16-node tile; loop over 16 column-tiles of H=256
__global__ __launch_bounds__(256) void k_forward(
    const float* __restrict__ aggQ, const float* __restrict__ ns,
    const float* __restrict__ nd,
    const float* __restrict__ W1, const float* __restrict__ b1,
    const float* __restrict__ W2p,
    float* __restrict__ ys, int nNodes, int nTiles) {
  __shared__ float T[8][16][17];
  const int wave = threadIdx.x >> 5;
  const int lane = threadIdx.x & 31;
  const int tile = blockIdx.x * 8 + wave;
  if (tile >= nTiles) return;
  const int base = tile * 16;
  const int n16 = lane & 15;
  const int hi = lane >> 4;
  const int kOff = hi * 2;
  int rowA = base + n16; if (rowA >= nNodes) rowA = nNodes - 1;
  const float ndv = nd[rowA];
  const float* aqr = aggQ + (size_t)rowA * 8;
  v2f a0, a1;                               // A fragments of aq = (A q), K=8
  a0.x = aqr[kOff + 0] * ndv; a0.y = aqr[kOff + 1] * ndv;
  a1.x = aqr[kOff + 4] * ndv; a1.y = aqr[kOff + 5] * ndv;
  v8f cy = v8splat(0.0f);                   // accumulates h1 @ W2 (16 x 16, cols 0-7 valid)
  float (*Tw)[17] = T[wave];
#pragma unroll 1
  for (int ct = 0; ct < 16; ++ct) {
    const int c0 = ct * 16;
    const int col = c0 + n16;
    v8f cz = v8splat(b1[col]);
    v2f b0, b1v;                            // B fragments of W1[:, col-tile]
    b0.x  = W1[(kOff + 0) * HH + col]; b0.y  = W1[(kOff + 1) * HH + col];
    b1v.x = W1[(kOff + 4) * HH + col]; b1v.y = W1[(kOff + 5) * HH + col];
    cz = wmma4(a0, b0, cz);
    cz = wmma4(a1, b1v, cz);
    __builtin_amdgcn_wave_barrier();
#pragma unroll
    for (int r = 0; r < 8; ++r)             // relu + transpose-stage in LDS
      Tw[r + hi * 8][n16] = fmaxf(cz[r], 0.0f);
    __builtin_amdgcn_wave_barrier();
#pragma unroll
    for (int kk = 0; kk < 4; ++kk) {        // accumulate h1_tile @ W2[rows c0..c0+15]
      const int kb = kk * 4 + kOff;
      v2f aT, bT;
      aT.x = Tw[n16][kb]; aT.y = Tw[n16][kb + 1];
      bT.x = W2p[(c0 + kb)     * 16 + n16];
      bT.y = W2p[(c0 + kb + 1) * 16 + n16];
      cy = wmma4(aT, bT, cy);
    }
    __builtin_amdgcn_wave_barrier();
  }
  if (n16 < 8) {
#pragma unroll
    for (int r = 0; r < 8; ++r) {
      const int node = base + r + hi * 8;
      if (node < nNodes)
        ys[(size_t)node * 8 + n16] = cy[r] * ns[node];  // pre-scale for A
    }
  }
}

// ---- h = aggY * norm_dst + b2 + q ------------------------------------------
__global__ void k_h(const float* __restrict__ aggY, const float* __restrict__ nd,
                    const float* __restrict__ b2, const float* __restrict__ q,
                    float* __restrict__ h, int n) {
  int i = blockIdx.x * blockDim.x + threadIdx.x;
  if (i < n * 8) {
    int node = i >> 3, d = i & 7;
    h[i] = aggY[i] * nd[node] + b2[d] + q[i];
  }
}

// ---- edge gravity force: G[src]+=g_e, G[dst]-=g_e ---------------------------
__global__ void k_force(const float* __restrict__ h, const float* __restrict__ md,
                        const int* __restrict__ src, const int* __restrict__ dst,
                        const float* __restrict__ grav,
                        float* __restrict__ G, int E) {
  int e = blockIdx.x * blockDim.x + threadIdx.x;
  if (e >= E) return;
  int s = src[e], d = dst[e];
  const float4* hs = (const float4*)(h + (size_t)s * 8);
  const float4* hd = (const float4*)(h + (size_t)d * 8);
  const float4* ms = (const float4*)(md + (size_t)s * 8);
  const float4* mdp = (const float4*)(md + (size_t)d * 8);
  float4 a0 = hs[0], a1 = hs[1], b0 = hd[0], b1 = hd[1];
  float4 m0 = ms[0], m1 = ms[1], n0 = mdp[0], n1 = mdp[1];
  float dx0 = a0.x - b0.x, dx1 = a0.y - b0.y, dx2 = a0.z - b0.z, dx3 = a0.w - b0.w;
  float dx4 = a1.x - b1.x, dx5 = a1.y - b1.y, dx6 = a1.z - b1.z, dx7 = a1.w - b1.w;
  float d2 = dx0*dx0 + dx1*dx1 + dx2*dx2 + dx3*dx3
           + dx4*dx4 + dx5*dx5 + dx6*dx6 + dx7*dx7;
  float S = m0.x*n0.x + m0.y*n0.y + m0.z*n0.z + m0.w*n0.w
          + m1.x*n1.x + m1.y*n1.y + m1.z*n1.z + m1.w*n1.w;
  float eu = sqrtf(d2);
  float coef = -0.5f * grav[0] * S / (d2 * eu);   // -c*S/euclid^3
  float* Gs = G + (size_t)s * 8;
  float* Gd = G + (size_t)d * 8;
  atomicAdd(Gs + 0, coef * dx0); atomicAdd(Gd + 0, -coef * dx0);
  atomicAdd(Gs + 1, coef * dx1); atomicAdd(Gd + 1, -coef * dx1);
  atomicAdd(Gs + 2, coef * dx2); atomicAdd(Gd + 2, -coef * dx2);
  atomicAdd(Gs + 3, coef * dx3); atomicAdd(Gd + 3, -coef * dx3);
  atomicAdd(Gs + 4, coef * dx4); atomicAdd(Gd + 4, -coef * dx4);
  atomicAdd(Gs + 5, coef * dx5); atomicAdd(Gd + 5, -coef * dx5);
  atomicAdd(Gs + 6, coef * dx6); atomicAdd(Gd + 6, -coef * dx6);
  atomicAdd(Gs + 7, coef * dx7); atomicAdd(Gd + 7, -coef * dx7);
}

// ---- backward fused: wn = (((A^T G) W2^T) ⊙ 1[z1>0]) W1^T  * norm_dst -------
__global__ __launch_bounds__(256) void k_backward(
    const float* __restrict__ aggQ, const float* __restrict__ aggG,
    const float* __restrict__ ns, const float* __restrict__ nd,
    const float* __restrict__ W1, const float* __restrict__ b1,
    const float* __restrict__ W2, const float* __restrict__ W1tp,
    float* __restrict__ wn, int nNodes, int nTiles) {
  __shared__ float T[8][16][17];
  const int wave = threadIdx.x >> 5;
  const int lane = threadIdx.x & 31;
  const int tile = blockIdx.x * 8 + wave;
  if (tile >= nTiles) return;
  const int base = tile * 16;
  const int n16 = lane & 15;
  const int hi = lane >> 4;
  const int kOff = hi * 2;
  int rowA = base + n16; if (rowA >= nNodes) rowA = nNodes - 1;
  const float ndv = nd[rowA];
  const float nsv = ns[rowA];
  const float* aqr = aggQ + (size_t)rowA * 8;
  const float* agr = aggG + (size_t)rowA * 8;
  v2f aq0, aq1, at0, at1;
  aq0.x = aqr[kOff + 0] * ndv; aq0.y = aqr[kOff + 1] * ndv;
  aq1.x = aqr[kOff + 4] * ndv; aq1.y = aqr[kOff + 5] * ndv;
  at0.x = agr[kOff + 0] * nsv; at0.y = agr[kOff + 1] * nsv;   // t = (A^T G)
  at1.x = agr[kOff + 4] * nsv; at1.y = agr[kOff + 5] * nsv;
  v8f cw = v8splat(0.0f);
  float (*Tw)[17] = T[wave];
#pragma unroll 1
  for (int ct = 0; ct < 16; ++ct) {
    const int c0 = ct * 16;
    const int col = c0 + n16;
    // v = t @ W2^T : B[k,n] = W2[(c0+n)*8 + k]
    v8f cv = v8splat(0.0f);
    v2f bv0, bv1;
    bv0.x = W2[col * 8 + kOff + 0]; bv0.y = W2[col * 8 + kOff + 1];
    bv1.x = W2[col * 8 + kOff + 4]; bv1.y = W2[col * 8 + kOff + 5];
    cv = wmma4(at0, bv0, cv);
    cv = wmma4(at1, bv1, cv);
    // recompute z1 tile for the relu mask
    v8f cz = v8splat(b1[col]);
    v2f b0, b1v;
    b0.x  = W1[(kOff + 0) * HH + col]; b0.y  = W1[(kOff + 1) * HH + col];
    b1v.x = W1[(kOff + 4) * HH + col]; b1v.y = W1[(kOff + 5) * HH + col];
    cz = wmma4(aq0, b0, cz);
    cz = wmma4(aq1, b1v, cz);
    __builtin_amdgcn_wave_barrier();
#pragma unroll
    for (int r = 0; r < 8; ++r)              // u = v ⊙ 1[z1>0], staged for transpose
      Tw[r + hi * 8][n16] = (cz[r] > 0.0f) ? cv[r] : 0.0f;
    __builtin_amdgcn_wave_barrier();
#pragma unroll
    for (int kk = 0; kk < 4; ++kk) {         // cw += u_tile @ W1^T[rows c0..c0+15]
      const int kb = kk * 4 + kOff;
      v2f aT, bT;
      aT.x = Tw[n16][kb]; aT.y = Tw[n16][kb + 1];
      bT.x = W1tp[(c0 + kb)     * 16 + n16];
      bT.y = W1tp[(c0 + kb + 1) * 16 + n16];
      cw = wmma4(aT, bT, cw);
    }
    __builtin_amdgcn_wave_barrier();
  }
  if (n16 < 8) {
#pragma unroll
    for (int r = 0; r < 8; ++r) {
      const int node = base + r + hi * 8;
      if (node < nNodes)
        wn[(size_t)node * 8 + n16] = cw[r] * nd[node];  // pre-scale for A^T
    }
  }
}

// ---- dHdQ = aggW * norm_src + G --------------------------------------------
__global__ void k_final(const float* __restrict__ aggW, const float* __restrict__ ns,
                        const float* __restrict__ G, float* __restrict__ out, int n) {
  int i = blockIdx.x * blockDim.x + threadIdx.x;
  if (i < n * 8) {
    int node = i >> 3, d = i & 7;
    out[(size_t)node * 16 + d] = aggW[i] * ns[node] + G[i];
  }
}

extern "C" void kernel_launch(void* const* d_in, const int* in_sizes, int n_in,
                              void* d_out, int out_size, void* d_ws, size_t ws_size,
                              hipStream_t stream) {
  const float* q   = (const float*)d_in[0];
  const float* p   = (const float*)d_in[1];
  const float* M   = (const float*)d_in[2];
  const int*   src = (const int*)d_in[3];
  const int*   dst = (const int*)d_in[4];
  const float* W1  = (const float*)d_in[5];
  const float* b1  = (const float*)d_in[6];
  const float* W2  = (const float*)d_in[7];
  const float* b2  = (const float*)d_in[8];
  const float* grv = (const float*)d_in[9];
  float* out = (float*)d_out;

  const int N = in_sizes[0] / 8;
  const int E = in_sizes[3];

  float* w = (float*)d_ws;
  float* outdeg = w; w += N;
  float* indeg  = w; w += N;
  float* aggQ = w; w += (size_t)8 * N;
  float* aggY = w; w += (size_t)8 * N;
  float* G    = w; w += (size_t)8 * N;
  float* aggG = w; w += (size_t)8 * N;
  float* aggW = w; w += (size_t)8 * N;   // --- zeroed region ends: 42N floats
  float* ns  = w; w += N;
  float* nd  = w; w += N;
  float* mdg = w; w += (size_t)8 * N;
  float* xs  = w; w += (size_t)8 * N;
  float* ys  = w; w += (size_t)8 * N;
  float* hb  = w; w += (size_t)8 * N;
  float* wn  = w; w += (size_t)8 * N;
  float* W2p  = w; w += 4096;            // padded [256][16]
  float* W1tp = w; w += 4096;            // padded transpose [256][16]

  hipMemsetAsync(d_ws, 0, (size_t)42 * N * sizeof(float), stream);

  const int TB = 256;
  const int gE  = (E + TB - 1) / TB;
  const int gN  = (N + TB - 1) / TB;
  const int gN8 = (8 * N + TB - 1) / TB;
  const int nTiles = (N + 15) / 16;
  const int gT = (nTiles + 7) / 8;

  k_padw<<<32, TB, 0, stream>>>(W1, W2, W2p, W1tp);
  k_deg<<<gE, TB, 0, stream>>>(src, dst, outdeg, indeg, E);
  k_prep<<<gN, TB, 0, stream>>>(q, p, M, outdeg, indeg, ns, nd, mdg, xs, out, N);
  k_edge_agg<<<gE, TB, 0, stream>>>(xs, src, dst, (const float*)nullptr, aggQ, E);
  k_forward<<<gT, TB, 0, stream>>>(aggQ, ns, nd, W1, b1, W2p, ys, N, nTiles);
  k_edge_agg<<<gE, TB, 0, stream>>>(ys, src, dst, (const float*)nullptr, aggY, E);
  k_h<<<gN8, TB, 0, stream>>>(aggY, nd, b2, q, hb, N);
  k_force<<<gE, TB, 0, stream>>>(hb, mdg, src, dst, grv, G, E);
  k_edge_agg<<<gE, TB, 0, stream>>>(G, dst, src, nd, aggG, E);
  k_backward<<<gT, TB, 0, stream>>>(aggQ, aggG, ns, nd, W1, b1, W2, W1tp, wn, N, nTiles);
  k_edge_agg<<<gE, TB, 0, stream>>>(wn, dst, src, (const float*)nullptr, aggW, E);
  k_final<<<gN8, TB, 0, stream>>>(aggW, ns, G, out, N);
}